// GNN_52063593562621
// MI455X (gfx1250) — compile-verified
//
#include <hip/hip_runtime.h>
#include <hip/hip_bf16.h>

#define N_NODES 100000
#define NEG_SLOPE 0.2f

typedef __attribute__((ext_vector_type(2))) float v2f;
typedef __attribute__((ext_vector_type(8))) float v8f;
typedef __attribute__((ext_vector_type(4))) unsigned int u32x4;
typedef __attribute__((ext_vector_type(4))) int i32x4;
typedef __attribute__((ext_vector_type(8))) int i32x8;

// ---------------------------------------------------------------------------
// Stage W[K,64] (K*64 f32, <=32KB) into LDS.
// Preferred: Tensor Data Mover (tensor_load_to_lds, TENSORcnt) issued once by
// wave 0 (TDM ignores EXEC; branching other waves around it avoids duplicate
// DMAs). Fallbacks: async global->LDS copy (ASYNCcnt), then plain VALU copy.
// ---------------------------------------------------------------------------
__device__ __forceinline__ void stage_W_to_lds(const float* __restrict__ W,
                                               float* sW, int nElems, int tid) {
#if __has_builtin(__builtin_amdgcn_tensor_load_to_lds)
  if (tid < 32) {  // one wave issues the DMA
    const unsigned lds_off =
        (unsigned)(size_t)(__attribute__((address_space(3))) float*)sW;
    const unsigned long long ga = (unsigned long long)(size_t)W;  // global VA
    const unsigned n = (unsigned)nElems;                          // <= 8192
    // D# group 0 (ISA 8.3): count=1 | lds_addr | global_addr[56:0] | type=2
    u32x4 g0 = {1u,
                lds_off,
                (unsigned)(ga & 0xffffffffu),
                (unsigned)((ga >> 32) & 0x01ffffffu) | (2u << 30)};
    // D# group 1 (ISA 8.4): data_size=2 (4B); tensor_dim0=n, tensor_dim1=1,
    // tile_dim0=n, tile_dim1=1, tensor_dim0_stride=n, tensor_dim1_stride=n.
    i32x8 g1 = {(int)0x00020000u,                       // [17:16]=data_size
                (int)((n & 0xffffu) << 16),             // dim0 lo16 @bits63:48
                (int)(((n >> 16) & 0xffffu) | (1u << 16)),  // dim0 hi | dim1=1
                (int)((n & 0xffffu) << 16),             // tile_dim0 @127:112
                (int)1,                                 // tile_dim1=1 @143:128
                (int)n,                                 // dim0_stride lo32
                (int)((n & 0xffffu) << 16),             // stride hi16=0|dim1_stride lo
                (int)((n >> 16) & 0xffffu)};            // dim1_stride hi
    i32x4 z4 = {0, 0, 0, 0};
#if __has_include(<hip/amd_detail/amd_gfx1250_TDM.h>)
    i32x8 z8 = {0, 0, 0, 0, 0, 0, 0, 0};
    __builtin_amdgcn_tensor_load_to_lds(g0, g1, z4, z4, z8, 0);  // 6-arg TC
#else
    __builtin_amdgcn_tensor_load_to_lds(g0, g1, z4, z4, 0);      // 5-arg TC
#endif
#if __has_builtin(__builtin_amdgcn_s_wait_tensorcnt)
    __builtin_amdgcn_s_wait_tensorcnt(0);
#else
    asm volatile("s_wait_tensorcnt 0x0" ::: "memory");
#endif
  }
  __syncthreads();
#elif __has_builtin(__builtin_amdgcn_global_load_async_to_lds_b32)
  for (int i = tid; i < nElems; i += 256)
    __builtin_amdgcn_global_load_async_to_lds_b32(
        (const __attribute__((address_space(1))) void*)(W + i),
        (__attribute__((address_space(3))) void*)
            ((__attribute__((address_space(3))) float*)sW + i),
        0, 0);
#if __has_builtin(__builtin_amdgcn_s_wait_asynccnt)
  __builtin_amdgcn_s_wait_asynccnt(0);
#else
  asm volatile("s_wait_asynccnt 0x0" ::: "memory");
#endif
  __syncthreads();
#else
  for (int i = tid; i < nElems; i += 256) sW[i] = W[i];
  __syncthreads();
#endif
}

// ---------------------------------------------------------------------------
// GEMM: H[N,64] = X[N,K] @ W[K,64] using V_WMMA_F32_16X16X4_F32 (full fp32).
// Block = 256 threads = 8 waves; each wave owns one 16x16 output tile; block
// covers 32 rows x 64 cols. N=100000 divisible by 32 (3125 blocks).
// ---------------------------------------------------------------------------
__global__ __launch_bounds__(256) void gat_gemm_wmma(const float* __restrict__ X,
                                                     const float* __restrict__ W,
                                                     float* __restrict__ H,
                                                     int K) {
  __shared__ float sW[128 * 64];  // 32 KB max (K<=128); LDS is 320 KB/WGP
  const int tid = threadIdx.x;
  stage_W_to_lds(W, sW, K * 64, tid);

  const int wave  = tid >> 5;
  const int lane  = tid & 31;
  const int m0    = (blockIdx.x * 2 + (wave >> 2)) * 16;
  const int n0    = (wave & 3) * 16;
  const int half  = lane >> 4;       // 0 or 1
  const int l15   = lane & 15;
  const int khalf = half * 2;

  v8f c = {};
  const float* xrow = X + (size_t)(m0 + l15) * K;
  for (int k0 = 0; k0 < K; k0 += 4) {
    v2f a = *(const v2f*)(xrow + k0 + khalf);         // A[row][k0+khalf..+1]
    v2f b;
    b.x = sW[(k0 + khalf) * 64 + n0 + l15];           // B[k0+khalf][col]
    b.y = sW[(k0 + khalf + 1) * 64 + n0 + l15];       // B[k0+khalf+1][col]
    c = __builtin_amdgcn_wmma_f32_16x16x4_f32(false, a, false, b, (short)0, c,
                                              false, false);
  }
  float* hp = H + (size_t)(m0 + half * 8) * 64 + n0 + l15;
#pragma unroll
  for (int r = 0; r < 8; ++r) hp[(size_t)r * 64] = c[r];
}

// ---------------------------------------------------------------------------
// Per-node attention coefficients: al_s[i] = <h[i], a_src>, al_d[i] = <h[i], a_dst>
// One wave32 per node; lane handles 2 dims (float2), shuffle tree reduction.
// ---------------------------------------------------------------------------
__global__ __launch_bounds__(256) void gat_attn(const float* __restrict__ H,
                                                const float* __restrict__ a_src,
                                                const float* __restrict__ a_dst,
                                                float* __restrict__ als,
                                                float* __restrict__ ald) {
  const int node = blockIdx.x * 8 + (threadIdx.x >> 5);
  if (node >= N_NODES) return;
  const int lane = threadIdx.x & 31;
  v2f hv  = *(const v2f*)(H + (size_t)node * 64 + lane * 2);
  v2f asv = *(const v2f*)(a_src + lane * 2);
  v2f adv = *(const v2f*)(a_dst + lane * 2);
  float ss = hv.x * asv.x + hv.y * asv.y;
  float dd = hv.x * adv.x + hv.y * adv.y;
#pragma unroll
  for (int off = 16; off > 0; off >>= 1) {
    ss += __shfl_xor(ss, off, 32);
    dd += __shfl_xor(dd, off, 32);
  }
  if (lane == 0) { als[node] = ss; ald[node] = dd; }
}

// ---------------------------------------------------------------------------
// Init segment-max (-inf), segment-sum (0), and output accumulator (0).
// ---------------------------------------------------------------------------
__global__ void gat_init(float* __restrict__ acc, float* __restrict__ m,
                         float* __restrict__ s) {
  const size_t i = (size_t)blockIdx.x * blockDim.x + threadIdx.x;
  if (i < (size_t)N_NODES * 64) acc[i] = 0.0f;
  if (i < N_NODES) { m[i] = -__builtin_huge_valf(); s[i] = 0.0f; }
}

__device__ __forceinline__ float leaky(float v) {
  return v > 0.0f ? v : NEG_SLOPE * v;
}

// Ordered-float atomic max via int/uint monotonic trick (init = -inf).
__device__ __forceinline__ void atomicMaxF(float* addr, float value) {
  if (value >= 0.0f) atomicMax((int*)addr, __float_as_int(value));
  else               atomicMin((unsigned int*)addr, __float_as_uint(value));
}

__device__ __forceinline__ void edge_ends(const long long* __restrict__ ei,
                                          int e, int E, long long& s_, long long& d_) {
  if (e < E) { s_ = ei[e]; d_ = ei[(size_t)E + e]; }
  else       { s_ = d_ = (long long)(e - E); }  // virtual self-loop
}

// Pass 1: segment max of leaky_relu scores over destinations.
__global__ void gat_edge_max(const long long* __restrict__ ei, int E, int Etot,
                             const float* __restrict__ als,
                             const float* __restrict__ ald,
                             float* __restrict__ m) {
  const int e = blockIdx.x * blockDim.x + threadIdx.x;
  if (e >= Etot) return;
  long long s_, d_; edge_ends(ei, e, E, s_, d_);
  atomicMaxF(m + d_, leaky(als[s_] + ald[d_]));
}

// Pass 2: segment sum of exp(score - max).
__global__ void gat_edge_sum(const long long* __restrict__ ei, int E, int Etot,
                             const float* __restrict__ als,
                             const float* __restrict__ ald,
                             const float* __restrict__ m,
                             float* __restrict__ ssum) {
  const int e = blockIdx.x * blockDim.x + threadIdx.x;
  if (e >= Etot) return;
  long long s_, d_; edge_ends(ei, e, E, s_, d_);
  const float sc = leaky(als[s_] + ald[d_]);
  atomicAdd(ssum + d_, expf(sc - m[d_]));
}

// Pass 3: weighted scatter. One wave per edge; lane handles 2 dims.
__global__ __launch_bounds__(256) void gat_edge_scatter(
    const long long* __restrict__ ei, int E, int Etot,
    const float* __restrict__ als, const float* __restrict__ ald,
    const float* __restrict__ m, const float* __restrict__ ssum,
    const float* __restrict__ H, float* __restrict__ acc) {
  const int e = blockIdx.x * 8 + (threadIdx.x >> 5);
  if (e >= Etot) return;
  const int lane = threadIdx.x & 31;
  long long s_, d_; edge_ends(ei, e, E, s_, d_);
  const float sc    = leaky(als[s_] + ald[d_]);
  const float alpha = expf(sc - m[d_]) / ssum[d_];
  v2f hv = *(const v2f*)(H + (size_t)s_ * 64 + lane * 2);
  float* ap = acc + (size_t)d_ * 64 + lane * 2;
  atomicAdd(ap,     alpha * hv.x);
  atomicAdd(ap + 1, alpha * hv.y);
}

// Bias add + optional ReLU into next-layer input (or final output).
__global__ void gat_finalize(const float* __restrict__ acc,
                             const float* __restrict__ b,
                             float* __restrict__ out, int do_relu) {
  const size_t i = (size_t)blockIdx.x * blockDim.x + threadIdx.x;
  if (i >= (size_t)N_NODES * 64) return;
  float v = acc[i] + b[i & 63];
  if (do_relu) v = fmaxf(v, 0.0f);
  out[i] = v;
}

// ---------------------------------------------------------------------------
// d_in: 0:x[N,128] 1:edge_index[2,E] (int64) then (W,as,ad,b) x 3 layers
// ---------------------------------------------------------------------------
extern "C" void kernel_launch(void* const* d_in, const int* in_sizes, int n_in,
                              void* d_out, int out_size, void* d_ws, size_t ws_size,
                              hipStream_t stream) {
  const float*     x0 = (const float*)d_in[0];
  const long long* ei = (const long long*)d_in[1];
  const int E    = in_sizes[1] / 2;
  const int Etot = E + N_NODES;
  const int ND   = N_NODES * 64;

  float* ws  = (float*)d_ws;
  float* h   = ws;                          // N*64
  float* acc = h   + (size_t)ND;            // N*64
  float* xb  = acc + (size_t)ND;            // N*64  (inter-layer activations)
  float* als = xb  + (size_t)ND;            // N
  float* ald = als + N_NODES;               // N
  float* mb  = ald + N_NODES;               // N
  float* sb  = mb  + N_NODES;               // N

  for (int layer = 0; layer < 3; ++layer) {
    const float* X   = (layer == 0) ? x0 : xb;
    const int    K   = (layer == 0) ? 128 : 64;
    const float* W   = (const float*)d_in[2 + 4 * layer];
    const float* as_ = (const float*)d_in[3 + 4 * layer];
    const float* ad_ = (const float*)d_in[4 + 4 * layer];
    const float* bb  = (const float*)d_in[5 + 4 * layer];
    float* dst = (layer == 2) ? (float*)d_out : xb;

    gat_gemm_wmma<<<N_NODES / 32, 256, 0, stream>>>(X, W, h, K);
    gat_attn<<<(N_NODES + 7) / 8, 256, 0, stream>>>(h, as_, ad_, als, ald);
    gat_init<<<(ND + 255) / 256, 256, 0, stream>>>(acc, mb, sb);
    gat_edge_max<<<(Etot + 255) / 256, 256, 0, stream>>>(ei, E, Etot, als, ald, mb);
    gat_edge_sum<<<(Etot + 255) / 256, 256, 0, stream>>>(ei, E, Etot, als, ald, mb, sb);
    gat_edge_scatter<<<(Etot + 7) / 8, 256, 0, stream>>>(ei, E, Etot, als, ald,
                                                         mb, sb, h, acc);
    gat_finalize<<<(ND + 255) / 256, 256, 0, stream>>>(acc, bb, dst,
                                                       (layer < 2) ? 1 : 0);
  }
}